// GCN_83013127897500
// MI455X (gfx1250) — compile-verified
//
#include <hip/hip_runtime.h>
#include <hip/hip_bf16.h>

typedef __attribute__((ext_vector_type(2))) float v2f;
typedef __attribute__((ext_vector_type(8))) float v8f;

// ---------------- elementwise helpers ----------------

__global__ void k_zero_f4(float4* __restrict__ p, long long n4) {
    long long i = (long long)blockIdx.x * blockDim.x + threadIdx.x;
    if (i < n4) p[i] = make_float4(0.f, 0.f, 0.f, 0.f);
}

__global__ void k_deg_init(float* __restrict__ deg, int n) {
    int i = blockIdx.x * blockDim.x + threadIdx.x;
    if (i < n) deg[i] = 1.0f;  // self-loop contributes 1
}

__global__ void k_deg_count(const long long* __restrict__ ei, float* __restrict__ deg,
                            long long E) {
    long long i = (long long)blockIdx.x * blockDim.x + threadIdx.x;
    if (i < E) unsafeAtomicAdd(deg + (int)ei[E + i], 1.0f);
}

__global__ void k_deg_to_dis(float* __restrict__ deg, int n) {
    int i = blockIdx.x * blockDim.x + threadIdx.x;
    if (i < n) deg[i] = rsqrtf(deg[i]);  // deg >= 1 always
}

// out[i,f] = agg[i,f] + xw[i,f]*dis[i]^2 + bias[f]  (optional relu)
__global__ void k_finalize(const float* __restrict__ xw, const float* __restrict__ dis,
                           const float* __restrict__ bias, float* __restrict__ out,
                           long long nf, int do_relu) {
    long long i = (long long)blockIdx.x * blockDim.x + threadIdx.x;
    if (i >= nf) return;
    int node = (int)(i >> 7);
    int f = (int)(i & 127);
    float di = dis[node];
    float v = out[i] + xw[i] * (di * di) + bias[f];
    out[i] = do_relu ? fmaxf(v, 0.f) : v;
}

// ---------------- sparse aggregation: one wave per edge ----------------
// lane L handles features [4L, 4L+4): coalesced 512B gather + 4 f32 atomics.
__global__ void __launch_bounds__(256)
k_scatter(const float* __restrict__ xw, const float* __restrict__ dis,
          const long long* __restrict__ ei, float* __restrict__ out, long long E) {
    int lane = threadIdx.x & 31;
    long long e = (long long)blockIdx.x * 8 + (threadIdx.x >> 5);
    if (e >= E) return;
    int s = (int)ei[e];
    int d = (int)ei[E + e];
    float norm = dis[s] * dis[d];
    float4 v = ((const float4*)(xw + (size_t)s * 128))[lane];
    float* o = out + (size_t)d * 128 + lane * 4;
    unsafeAtomicAdd(o + 0, v.x * norm);
    unsafeAtomicAdd(o + 1, v.y * norm);
    unsafeAtomicAdd(o + 2, v.z * norm);
    unsafeAtomicAdd(o + 3, v.w * norm);
}

// ---------------- dense 128x128 transform via fp32 WMMA ----------------
// C[M,128] = A[M,128] @ B[128,128]; block = 8 waves; wave w -> cols [16w,16w+16)
__global__ void __launch_bounds__(256)
k_gemm128(const float* __restrict__ A, const float* __restrict__ B,
          float* __restrict__ C) {
    __shared__ float As[16 * 128];
    int t = threadIdx.x;
    int row0 = blockIdx.x * 16;
    // stage the 16x128 A tile (coalesced)
    for (int i = t; i < 16 * 128; i += 256)
        As[i] = A[(size_t)(row0 + (i >> 7)) * 128 + (i & 127)];
    __syncthreads();

    int lane = t & 31;
    int col0 = (t >> 5) * 16;
    int lh = lane & 15;         // A: row M ; B/D: col N
    int kp = (lane >> 4) * 2;   // K sub-pair within step-of-4

    v8f acc = {};
    for (int k0 = 0; k0 < 128; k0 += 4) {
        v2f a, b;
        a.x = As[lh * 128 + k0 + kp];
        a.y = As[lh * 128 + k0 + kp + 1];
        b.x = B[(size_t)(k0 + kp) * 128 + col0 + lh];
        b.y = B[(size_t)(k0 + kp + 1) * 128 + col0 + lh];
        acc = __builtin_amdgcn_wmma_f32_16x16x4_f32(false, a, false, b,
                                                    (short)0, acc, false, false);
    }
    int rbase = row0 + 8 * (lane >> 4);  // D: VGPR v holds row v + 8*(lane>>4)
    for (int v = 0; v < 8; ++v)
        C[(size_t)(rbase + v) * 128 + col0 + lh] = acc[v];
}

// ---------------- classifier head: [N,128] @ [128,10] + bf (N padded to 16) ----
__global__ void __launch_bounds__(32)
k_logits(const float* __restrict__ H, const float* __restrict__ Wf,
         const float* __restrict__ bf, float* __restrict__ out) {
    int lane = threadIdx.x;
    int row0 = blockIdx.x * 16;
    int lh = lane & 15;
    int kp = (lane >> 4) * 2;

    v8f acc = {};
    for (int k0 = 0; k0 < 128; k0 += 4) {
        v2f a, b;
        a.x = H[(size_t)(row0 + lh) * 128 + k0 + kp];
        a.y = H[(size_t)(row0 + lh) * 128 + k0 + kp + 1];
        b.x = (lh < 10) ? Wf[(k0 + kp) * 10 + lh] : 0.f;
        b.y = (lh < 10) ? Wf[(k0 + kp + 1) * 10 + lh] : 0.f;
        acc = __builtin_amdgcn_wmma_f32_16x16x4_f32(false, a, false, b,
                                                    (short)0, acc, false, false);
    }
    if (lh < 10) {
        float bias = bf[lh];
        int rbase = row0 + 8 * (lane >> 4);
        for (int v = 0; v < 8; ++v)
            out[(size_t)(rbase + v) * 10 + lh] = acc[v] + bias;
    }
}

// ---------------- driver ----------------

extern "C" void kernel_launch(void* const* d_in, const int* in_sizes, int n_in,
                              void* d_out, int out_size, void* d_ws, size_t ws_size,
                              hipStream_t stream) {
    const float* x  = (const float*)d_in[0];
    const float* W1 = (const float*)d_in[1];
    const float* b1 = (const float*)d_in[2];
    const float* W2 = (const float*)d_in[3];
    const float* b2 = (const float*)d_in[4];
    const float* Wf = (const float*)d_in[5];
    const float* bf = (const float*)d_in[6];
    const long long* ei = (const long long*)d_in[7];

    const int N = in_sizes[0] / 128;            // 100000 (multiple of 16)
    const long long E = in_sizes[7] / 2;        // 1,600,000
    const long long NF = (long long)N * 128;

    float* ws   = (float*)d_ws;
    float* dis  = ws;                            // [N]
    float* bufA = ws + 102400;                   // xw scratch [N,128]
    float* bufB = bufA + (size_t)N * 128;        // h1 [N,128]
    float* h_out  = (float*)d_out;               // h  [N,128]
    float* logits = h_out + (size_t)N * 128;     // [N,10]

    const int TB = 256;
    dim3 blk(TB);
    int gN   = (N + TB - 1) / TB;
    long long gE   = (E + TB - 1) / TB;
    long long gNF  = (NF + TB - 1) / TB;
    long long gNF4 = (NF / 4 + TB - 1) / TB;
    long long gEw  = (E + 7) / 8;                // one wave per edge
    int gTile = N / 16;                          // 6250

    // degrees -> deg^-1/2
    k_deg_init<<<gN, blk, 0, stream>>>(dis, N);
    k_deg_count<<<(unsigned)gE, blk, 0, stream>>>(ei, dis, E);
    k_deg_to_dis<<<gN, blk, 0, stream>>>(dis, N);

    // zero aggregation targets (harness poisons buffers)
    k_zero_f4<<<(unsigned)gNF4, blk, 0, stream>>>((float4*)bufB, NF / 4);
    k_zero_f4<<<(unsigned)gNF4, blk, 0, stream>>>((float4*)h_out, NF / 4);

    // layer 1: xw1 = x @ W1 ; aggregate ; + self-loop + b1 ; relu
    k_gemm128<<<gTile, blk, 0, stream>>>(x, W1, bufA);
    k_scatter<<<(unsigned)gEw, blk, 0, stream>>>(bufA, dis, ei, bufB, E);
    k_finalize<<<(unsigned)gNF, blk, 0, stream>>>(bufA, dis, b1, bufB, NF, 1);

    // layer 2: xw2 = h1 @ W2 ; aggregate into d_out ; + self-loop + b2
    k_gemm128<<<gTile, blk, 0, stream>>>(bufB, W2, bufA);
    k_scatter<<<(unsigned)gEw, blk, 0, stream>>>(bufA, dis, ei, h_out, E);
    k_finalize<<<(unsigned)gNF, blk, 0, stream>>>(bufA, dis, b2, h_out, NF, 0);

    // classifier head
    k_logits<<<gTile, dim3(32), 0, stream>>>(h_out, Wf, bf, logits);
}